// Discriminator_80564996538506
// MI455X (gfx1250) — compile-verified
//
#include <hip/hip_runtime.h>
#include <hip/hip_bf16.h>
#include <math.h>

typedef __attribute__((ext_vector_type(2))) float v2f;
typedef __attribute__((ext_vector_type(8))) float v8f;

#define N_NODES   50000
#define N_EDGES   800000
#define HID       64
#define PRE_IN    144     // 64 + 64 + 16
#define POST_IN   768     // 12 * HID
#define EPS_F     1e-5f
#define FLT_MAX_F 3.402823466e+38f

#define E_TILES   (N_EDGES/16)   // 50000 edge tiles of 16
#define V_TILES   (N_NODES/16)   // 3125 node tiles of 16

// ---------------- init workspace ----------------
__global__ __launch_bounds__(256) void k_init(float* deg, float* asum, float* asq,
                                              int* amx, int* amn, float* scal) {
  int i = blockIdx.x * blockDim.x + threadIdx.x;
  if (i < N_NODES * HID) {
    asum[i] = 0.f;
    asq[i]  = 0.f;
    amx[i]  = __float_as_int(-FLT_MAX_F);   // monotone int compare valid: values >= 0
    amn[i]  = __float_as_int( FLT_MAX_F);
  }
  if (i < N_NODES) deg[i] = 0.f;
  if (i < 2)       scal[i] = 0.f;           // scal[0]=sum(log deg), scal[1]=bce sum
}

// ---------------- degree ----------------
__global__ __launch_bounds__(256) void k_deg(const int* __restrict__ edst, float* __restrict__ deg) {
  int i = blockIdx.x * blockDim.x + threadIdx.x;
  if (i < N_EDGES) atomicAdd(&deg[edst[i]], 1.f);
}

// ---------------- per-edge pre-MLP (WMMA f32) + scatter aggregation ----------------
#define EKW 2                 // waves per block
#define EKT (EKW*32)
#define TPW 8                 // edge-tiles per wave  -> blocks = E_TILES/(EKW*TPW) = 3125
#define WPS (PRE_IN + 1)      // padded W_pre row stride (145: odd vs 64 banks)

__global__ __launch_bounds__(EKT) void k_edge(
    const float* __restrict__ nf, const float* __restrict__ ef,
    const int* __restrict__ esrc, const int* __restrict__ edst,
    const float* __restrict__ W_pre, const float* __restrict__ b_pre,
    float* __restrict__ asum, float* __restrict__ asq,
    int* __restrict__ amx, int* __restrict__ amn) {
  __shared__ float sW[HID * WPS];          // 64x145 f32 = 37,120 B
  __shared__ float sM[EKW][16 * 145];      // 2 x 9,280 B
  const int tid = threadIdx.x;
  for (int i = tid; i < HID * PRE_IN; i += EKT) {
    int c = i / PRE_IN, k = i - c * PRE_IN;
    sW[c * WPS + k] = W_pre[i];
  }
  __syncthreads();

  const int wave = tid >> 5, lane = tid & 31;
  const int row = lane & 15, hi = lane >> 4;  // ISA A/B frag: lanes 0-15 K{0,1}, 16-31 K{2,3}
  float* mi = &sM[wave][0];

  float bp[4];
#pragma unroll
  for (int n = 0; n < 4; ++n) bp[n] = b_pre[n * 16 + row];

  for (int it = 0; it < TPW; ++it) {
    const int tile = (blockIdx.x * EKW + wave) * TPW + it;
    const int e0 = tile * 16;
    // stage m_in tile [16 x 144] = [h_dst | h_src | e] into LDS (coalesced gathers)
    for (int r = 0; r < 16; ++r) {
      const int d = edst[e0 + r], s = esrc[e0 + r];
      mi[r * 145 + lane]       = nf[d * HID + lane];
      mi[r * 145 + 32 + lane]  = nf[d * HID + 32 + lane];
      mi[r * 145 + 64 + lane]  = nf[s * HID + lane];
      mi[r * 145 + 96 + lane]  = nf[s * HID + 32 + lane];
      if (lane < 16) mi[r * 145 + 128 + lane] = ef[(e0 + r) * 16 + lane];
    }
    v8f acc[4] = {};
    for (int kc = 0; kc < PRE_IN / 4; ++kc) {        // 36 K-chunks of 4
      const int kb = kc * 4 + hi * 2;
      v2f a; a.x = mi[row * 145 + kb]; a.y = mi[row * 145 + kb + 1];
#pragma unroll
      for (int n = 0; n < 4; ++n) {                  // 4 output tiles of 16 -> 64 hidden
        const int col = n * 16 + row;                // B[k][n] = W_pre[n][k]
        v2f b; b.x = sW[col * WPS + kb]; b.y = sW[col * WPS + kb + 1];
        acc[n] = __builtin_amdgcn_wmma_f32_16x16x4_f32(
            false, a, false, b, (short)0, acc[n], false, false);
      }
    }
    // epilogue: bias + relu, then 4 atomics per element into dst-node aggregators
#pragma unroll
    for (int n = 0; n < 4; ++n) {
      const int col = n * 16 + row;
#pragma unroll
      for (int r = 0; r < 8; ++r) {
        const int m = r + hi * 8;                    // C layout: VGPR r -> M=r (lanes<16) / r+8
        const int node = edst[e0 + m];
        const float v = fmaxf(acc[n][r] + bp[n], 0.f);
        const int o = node * HID + col;
        atomicAdd(&asum[o], v);
        atomicAdd(&asq[o], v * v);
        atomicMax(&amx[o], __float_as_int(v));       // v >= 0 -> int order == float order
        atomicMin(&amn[o], __float_as_int(v));
      }
    }
  }
}

// ---------------- delta = sum(log(deg_c+1)) with block pre-reduction ----------------
__global__ __launch_bounds__(256) void k_delta(const float* __restrict__ deg, float* __restrict__ scal) {
  __shared__ float red[256];
  const int i = blockIdx.x * 256 + threadIdx.x;
  float v = 0.f;
  if (i < N_NODES) v = logf(fmaxf(deg[i], 1.f) + 1.f);
  red[threadIdx.x] = v;
  __syncthreads();
  for (int s = 128; s > 0; s >>= 1) {
    if (threadIdx.x < s) red[threadIdx.x] += red[threadIdx.x + s];
    __syncthreads();
  }
  if (threadIdx.x == 0) atomicAdd(&scal[0], red[0]);
}

// ---------------- per-node post-MLP (WMMA f32) + logits + BCE ----------------
#define NKW 2
#define NKT (NKW*32)
#define AGS 259               // padded agg row stride (3 mod 64 -> conflict-free frag reads)

__global__ __launch_bounds__(NKT) void k_node(
    const float* __restrict__ deg, const float* __restrict__ asum, const float* __restrict__ asq,
    const int* __restrict__ amx, const int* __restrict__ amn,
    const float* __restrict__ W_post, const float* __restrict__ b_post,
    const float* __restrict__ W_o, const float* __restrict__ b_o,
    const float* __restrict__ mask, const float* __restrict__ scal,
    float* __restrict__ bce) {
  __shared__ float sA[NKW][16 * AGS];      // 2 x 16,576 B
  const int tid = threadIdx.x, wave = tid >> 5, lane = tid & 31;
  const int tile = blockIdx.x * NKW + wave;
  if (tile >= V_TILES) return;             // wave-uniform -> EXEC all-ones where WMMA runs
  const float delta = scal[0] * (1.f / (float)N_NODES);
  const int v0 = tile * 16;
  float* ag = &sA[wave][0];
  const int row = lane & 15, hi = lane >> 4;

  // build agg tile [16 x 256] = [mean | min | max | std]
  for (int r = 0; r < 16; ++r) {
    const int v = v0 + r;
    const float d = deg[v];
    const float inv = 1.f / fmaxf(d, 1.f);
    const bool has = d > 0.f;
    for (int c = lane; c < HID; c += 32) {
      const int o = v * HID + c;
      const float s = asum[o], q = asq[o];
      const float mean = s * inv;
      const float var = q * inv - mean * mean;
      ag[r * AGS + c]        = mean;
      ag[r * AGS + 64 + c]   = has ? __int_as_float(amn[o]) : 0.f;
      ag[r * AGS + 128 + c]  = has ? __int_as_float(amx[o]) : 0.f;
      ag[r * AGS + 192 + c]  = sqrtf(fmaxf(var, 0.f) + EPS_F);
    }
  }
  // per-lane degree scalers for this lane's A-row (node v0+row)
  float amp, att;
  {
    const float ld = logf(fmaxf(deg[v0 + row], 1.f) + 1.f);
    amp = ld / delta;
    att = delta / ld;
  }
  v8f acc[4] = {};
  for (int kc = 0; kc < POST_IN / 4; ++kc) {         // 192 K-chunks (3 x 256)
    const int K = kc * 4 + hi * 2;
    const int sect = K >> 8;                         // 0: identity, 1: amp, 2: att
    const float sc = (sect == 0) ? 1.f : ((sect == 1) ? amp : att);
    const int kb = K & 255;
    v2f a; a.x = sc * ag[row * AGS + kb]; a.y = sc * ag[row * AGS + kb + 1];
#pragma unroll
    for (int n = 0; n < 4; ++n) {
      const int col = n * 16 + row;                  // B[k][n] = W_post[n][k]
      v2f b; b.x = W_post[col * POST_IN + K]; b.y = W_post[col * POST_IN + K + 1];
      acc[n] = __builtin_amdgcn_wmma_f32_16x16x4_f32(
          false, a, false, b, (short)0, acc[n], false, false);
    }
  }
  // logits: per node, its 64 h-values live in 16 lanes x 4 n-tiles for a fixed r
  float wo[4], bpo[4];
#pragma unroll
  for (int n = 0; n < 4; ++n) { wo[n] = W_o[n * 16 + row]; bpo[n] = b_post[n * 16 + row]; }
  float bl = 0.f;
#pragma unroll
  for (int r = 0; r < 8; ++r) {
    float p = 0.f;
#pragma unroll
    for (int n = 0; n < 4; ++n)
      p += fmaxf(acc[n][r] + bpo[n], 0.f) * wo[n];
    p += __shfl_xor(p, 1); p += __shfl_xor(p, 2); p += __shfl_xor(p, 4); p += __shfl_xor(p, 8);
    if (row == 0) {
      const int node = v0 + r + hi * 8;
      const float z = p + b_o[0];
      const float y = mask[node];
      bl += fmaxf(z, 0.f) - z * y + log1pf(expf(-fabsf(z)));
    }
  }
  bl += __shfl_xor(bl, 16);
  if (lane == 0) atomicAdd(bce, bl);
}

// ---------------- finalize ----------------
__global__ void k_fin(const float* __restrict__ scal, float* __restrict__ out) {
  out[0] = scal[1] * (1.f / (float)N_NODES);
}

extern "C" void kernel_launch(void* const* d_in, const int* in_sizes, int n_in,
                              void* d_out, int out_size, void* d_ws, size_t ws_size,
                              hipStream_t stream) {
  (void)in_sizes; (void)n_in; (void)out_size; (void)ws_size;
  const float* nf     = (const float*)d_in[0];
  const float* ef     = (const float*)d_in[1];
  const int*   esrc   = (const int*)  d_in[2];
  const int*   edst   = (const int*)  d_in[3];
  const float* mask   = (const float*)d_in[4];
  const float* W_pre  = (const float*)d_in[5];
  const float* b_pre  = (const float*)d_in[6];
  const float* W_post = (const float*)d_in[7];
  const float* b_post = (const float*)d_in[8];
  const float* W_o    = (const float*)d_in[9];
  const float* b_o    = (const float*)d_in[10];

  // workspace layout (floats): deg[N] | sum[N*64] | sq[N*64] | max[N*64] | min[N*64] | scal[2]
  float* ws   = (float*)d_ws;
  float* deg  = ws;
  float* asum = deg + N_NODES;
  float* asq  = asum + N_NODES * HID;
  int*   amx  = (int*)(asq + N_NODES * HID);
  int*   amn  = amx + N_NODES * HID;
  float* scal = (float*)(amn + N_NODES * HID);
  float* out  = (float*)d_out;

  k_init <<<(N_NODES * HID + 255) / 256, 256, 0, stream>>>(deg, asum, asq, amx, amn, scal);
  k_deg  <<<(N_EDGES + 255) / 256, 256, 0, stream>>>(edst, deg);
  k_edge <<<E_TILES / (EKW * TPW), EKT, 0, stream>>>(nf, ef, esrc, edst, W_pre, b_pre,
                                                     asum, asq, amx, amn);
  k_delta<<<(N_NODES + 255) / 256, 256, 0, stream>>>(deg, scal);
  k_node <<<(V_TILES + NKW - 1) / NKW, NKT, 0, stream>>>(deg, asum, asq, amx, amn,
                                                         W_post, b_post, W_o, b_o,
                                                         mask, scal, scal + 1);
  k_fin  <<<1, 1, 0, stream>>>(scal, out);
}